// GAT_12867722019099
// MI455X (gfx1250) — compile-verified
//
#include <hip/hip_runtime.h>
#include <hip/hip_bf16.h>

// ---------------------------------------------------------------------------
// GAT forward on gfx1250 (CDNA5):
//   h = x @ W                         -> WMMA f32 16x16x4 GEMM (double-buffered)
//   a_src/a_dst per-node logits       -> wave-per-head dot products
//   edge pass 1: leaky-relu score + segment-max (encoded u32 atomicMax)
//   edge pass 2: exp, denom atomicAdd, un-normalized weighted scatter-add
//   finalize: divide by denom, mean over heads, + bias
// ---------------------------------------------------------------------------

#define GAT_N   50000
#define GAT_E   200000
#define GAT_H   8
#define GAT_C   256
#define GAT_IN  256
#define GAT_HC  (GAT_H * GAT_C)     // 2048
#define NEG_SLOPE 0.2f
#define GAT_EPS 1e-16f

typedef __attribute__((ext_vector_type(2))) float v2f;
typedef __attribute__((ext_vector_type(8))) float v8f;

// ---- monotonic float <-> u32 order-preserving encoding (for atomicMax) ----
__device__ __forceinline__ unsigned fenc(float x) {
    unsigned u = __float_as_uint(x);
    return (u & 0x80000000u) ? ~u : (u | 0x80000000u);
}
__device__ __forceinline__ float fdec(unsigned e) {
    return (e & 0x80000000u) ? __uint_as_float(e & 0x7FFFFFFFu)
                             : __uint_as_float(~e);
}

// ---------------------------------------------------------------------------
// Kernel 1: h[N,2048] = x[N,256] @ W[256,2048], f32 WMMA 16x16x4.
// Block = 256 threads (8 waves). Block tile 128(M) x 64(N), K staged by 32.
// Wave w owns rows [16w,16w+16) and all 4 n-subtiles (16 cols each).
//
// A LDS: row-major, stride 34 (gcd-picked so half-wave k-offset-2 reads hit
//        distinct banks); (a.x,a.y) contiguous -> ds_load_b64.
// B LDS: k-pair interleaved: element (k,n) at (k>>1)*160 + n*2 + (k&1), so
//        (b.x,b.y)=(B[k][n],B[k+1][n]) is one aligned ds_load_b64; pair
//        stride 160 mod 64 == 32 puts the upper half-wave (k+2 -> next pair
//        row) on the other 32 banks -> conflict-free.
// Global loads for block k+1 are prefetched into registers while block k's
// WMMAs run (register double buffering).
// ---------------------------------------------------------------------------
#define AS_STRIDE 34
#define BS_PSTRIDE 160   // dwords per k-pair row: 64 cols * 2 + 32 pad

__global__ __launch_bounds__(256)
void gat_gemm_wmma(const float* __restrict__ x, const float* __restrict__ W,
                   float* __restrict__ h) {
    __shared__ float As[128 * AS_STRIDE];
    __shared__ float Bs[16 * BS_PSTRIDE];

    const int tid      = threadIdx.x;
    const int wave     = tid >> 5;
    const int lane     = tid & 31;
    const int laneHalf = lane & 15;
    const int off2     = (lane < 16) ? 0 : 2;   // K sub-offset per ISA layout

    const int n0 = blockIdx.x * 64;
    const int m0 = blockIdx.y * 128;

    v8f acc[4] = {};

    float4 aReg[4];
    float2 bReg[2][2];   // [iter][row-of-pair]

    // ---- register prefetch of one K-block ----
    auto loadRegs = [&](int k0) {
        #pragma unroll
        for (int i = 0; i < 4; ++i) {
            int fid = tid + i * 256;        // float4 id 0..1023 (A: 128x32)
            int row = fid >> 3;             // 8 float4 per row
            int col = (fid & 7) * 4;
            int gm  = m0 + row;
            aReg[i] = (gm < GAT_N)
                ? *(const float4*)(x + (size_t)gm * GAT_IN + k0 + col)
                : make_float4(0.f, 0.f, 0.f, 0.f);
        }
        #pragma unroll
        for (int i = 0; i < 2; ++i) {
            int fid   = tid + i * 256;      // write id 0..511 (B: 16 pairs x 32 npairs)
            int p     = fid >> 5;           // k-pair 0..15
            int ncol  = (fid & 31) * 2;     // n column (even)
            const float* w0 = W + (size_t)(k0 + 2 * p) * GAT_HC + n0 + ncol;
            bReg[i][0] = *(const float2*)(w0);
            bReg[i][1] = *(const float2*)(w0 + GAT_HC);
        }
    };
    // ---- commit prefetched registers to LDS ----
    auto storeRegs = [&]() {
        #pragma unroll
        for (int i = 0; i < 4; ++i) {
            int fid = tid + i * 256;
            int row = fid >> 3;
            int col = (fid & 7) * 4;
            float* a = &As[row * AS_STRIDE + col];
            a[0] = aReg[i].x; a[1] = aReg[i].y; a[2] = aReg[i].z; a[3] = aReg[i].w;
        }
        #pragma unroll
        for (int i = 0; i < 2; ++i) {
            int fid  = tid + i * 256;
            int p    = fid >> 5;
            int ncol = (fid & 31) * 2;
            // interleave: {B[2p][n], B[2p+1][n], B[2p][n+1], B[2p+1][n+1]}
            float4 v = make_float4(bReg[i][0].x, bReg[i][1].x,
                                   bReg[i][0].y, bReg[i][1].y);
            *(float4*)&Bs[p * BS_PSTRIDE + ncol * 2] = v;
        }
    };

    loadRegs(0);
    const int mrow = (wave * 16 + laneHalf) * AS_STRIDE;

    for (int k0 = 0; k0 < GAT_IN; k0 += 32) {
        storeRegs();
        __syncthreads();
        if (k0 + 32 < GAT_IN) loadRegs(k0 + 32);   // overlap with WMMAs below

        #pragma unroll
        for (int ks = 0; ks < 32; ks += 4) {
            v2f a = *(const v2f*)&As[mrow + ks + off2];
            const int bbase = ((ks + off2) >> 1) * BS_PSTRIDE + laneHalf * 2;
            #pragma unroll
            for (int nt = 0; nt < 4; ++nt) {
                v2f b = *(const v2f*)&Bs[bbase + nt * 32];
                acc[nt] = __builtin_amdgcn_wmma_f32_16x16x4_f32(
                    /*neg_a=*/false, a, /*neg_b=*/false, b,
                    /*c_mod=*/(short)0, acc[nt],
                    /*reuse_a=*/false, /*reuse_b=*/false);
            }
        }
        __syncthreads();
    }

    // --- store C: VGPR j holds M = (lane<16 ? j : 8+j), N = laneHalf ---
    const int mbase = m0 + wave * 16 + ((lane < 16) ? 0 : 8);
    #pragma unroll
    for (int nt = 0; nt < 4; ++nt) {
        const int gn = n0 + nt * 16 + laneHalf;
        #pragma unroll
        for (int j = 0; j < 8; ++j) {
            int gm = mbase + j;
            if (gm < GAT_N)
                h[(size_t)gm * GAT_HC + gn] = acc[nt][j];
        }
    }
}

// ---------------------------------------------------------------------------
// Kernel 2: per-node, per-head attention logits. Block = node, wave = head.
// ---------------------------------------------------------------------------
__global__ __launch_bounds__(256)
void gat_logits(const float* __restrict__ h,
                const float* __restrict__ att_src, const float* __restrict__ att_dst,
                float* __restrict__ a_src, float* __restrict__ a_dst) {
    const int n    = blockIdx.x;
    const int head = threadIdx.x >> 5;
    const int lane = threadIdx.x & 31;
    const float* hp = h + (size_t)n * GAT_HC + head * GAT_C;
    const float* as = att_src + head * GAT_C;
    const float* ad = att_dst + head * GAT_C;
    float s0 = 0.f, s1 = 0.f;
    #pragma unroll
    for (int c = lane; c < GAT_C; c += 32) {
        float v = hp[c];
        s0 += v * as[c];
        s1 += v * ad[c];
    }
    #pragma unroll
    for (int o = 16; o > 0; o >>= 1) {
        s0 += __shfl_xor(s0, o, 32);
        s1 += __shfl_xor(s1, o, 32);
    }
    if (lane == 0) {
        a_src[n * GAT_H + head] = s0;
        a_dst[n * GAT_H + head] = s1;
    }
}

// ---------------------------------------------------------------------------
// Kernel 3: edge pass 1 — leaky-relu score, segment-max via encoded atomicMax
// ---------------------------------------------------------------------------
__global__ __launch_bounds__(256)
void gat_edge_score(const int* __restrict__ ei,
                    const float* __restrict__ a_src, const float* __restrict__ a_dst,
                    float* __restrict__ score, unsigned* __restrict__ menc) {
    const int e = blockIdx.x * blockDim.x + threadIdx.x;
    if (e >= GAT_E) return;
    const int s = ei[e];
    const int d = ei[GAT_E + e];
    #pragma unroll
    for (int hh = 0; hh < GAT_H; ++hh) {
        float v = a_src[s * GAT_H + hh] + a_dst[d * GAT_H + hh];
        v = (v > 0.f) ? v : v * NEG_SLOPE;
        score[(size_t)e * GAT_H + hh] = v;
        atomicMax(&menc[d * GAT_H + hh], fenc(v));
    }
}

// ---------------------------------------------------------------------------
// Kernel 4: edge pass 2 — exp, denom accumulation, un-normalized weighted
// scatter-add of h[src] into acc[dst]. One 256-thread block per edge; thread
// t owns elements [8t, 8t+8) -> head = t>>5 is wave-uniform; 128-bit loads.
// ---------------------------------------------------------------------------
__global__ __launch_bounds__(256)
void gat_edge_scatter(const int* __restrict__ ei,
                      const float* __restrict__ score, const unsigned* __restrict__ menc,
                      const float* __restrict__ h,
                      float* __restrict__ acc, float* __restrict__ denom) {
    __shared__ float wexp[GAT_H];
    __shared__ int sS, sD;
    const int e = blockIdx.x;
    if (threadIdx.x == 0) { sS = ei[e]; sD = ei[GAT_E + e]; }
    __syncthreads();
    const int s = sS, d = sD;
    if (threadIdx.x < GAT_H) {
        const int hh = threadIdx.x;
        float m  = fdec(menc[d * GAT_H + hh]);
        float ex = __expf(score[(size_t)e * GAT_H + hh] - m);
        wexp[hh] = ex;
        atomicAdd(&denom[d * GAT_H + hh], ex);
    }
    __syncthreads();
    const int t    = threadIdx.x;
    const float w  = wexp[t >> 5];
    const size_t sb = (size_t)s * GAT_HC + t * 8;
    const size_t db = (size_t)d * GAT_HC + t * 8;
    float4 v0 = *(const float4*)(h + sb);
    float4 v1 = *(const float4*)(h + sb + 4);
    atomicAdd(acc + db + 0, v0.x * w);
    atomicAdd(acc + db + 1, v0.y * w);
    atomicAdd(acc + db + 2, v0.z * w);
    atomicAdd(acc + db + 3, v0.w * w);
    atomicAdd(acc + db + 4, v1.x * w);
    atomicAdd(acc + db + 5, v1.y * w);
    atomicAdd(acc + db + 6, v1.z * w);
    atomicAdd(acc + db + 7, v1.w * w);
}

// ---------------------------------------------------------------------------
// Kernel 5: finalize — divide by (denom + eps), mean over heads, add bias
// ---------------------------------------------------------------------------
__global__ __launch_bounds__(256)
void gat_finalize(const float* __restrict__ acc, const float* __restrict__ denom,
                  const float* __restrict__ bias, float* __restrict__ out) {
    __shared__ float inv[GAT_H];
    const int n = blockIdx.x;
    if (threadIdx.x < GAT_H)
        inv[threadIdx.x] = 1.0f / (denom[n * GAT_H + threadIdx.x] + GAT_EPS);
    __syncthreads();
    const int c = threadIdx.x;
    float sum = 0.f;
    #pragma unroll
    for (int hh = 0; hh < GAT_H; ++hh)
        sum += acc[(size_t)n * GAT_HC + hh * GAT_C + c] * inv[hh];
    out[n * GAT_C + c] = sum * (1.0f / GAT_H) + bias[c];
}

// ---------------------------------------------------------------------------
extern "C" void kernel_launch(void* const* d_in, const int* in_sizes, int n_in,
                              void* d_out, int out_size, void* d_ws, size_t ws_size,
                              hipStream_t stream) {
    const float* x        = (const float*)d_in[0];   // [N, 256]
    const int*   ei       = (const int*)  d_in[1];   // [2, E]
    const float* W        = (const float*)d_in[2];   // [256, 2048]
    const float* att_src  = (const float*)d_in[3];   // [8, 256]
    const float* att_dst  = (const float*)d_in[4];   // [8, 256]
    const float* bias     = (const float*)d_in[5];   // [256]
    float* out = (float*)d_out;

    // ---- workspace layout (floats) ----
    float* ws    = (float*)d_ws;
    float* h     = ws;                                    // N*2048
    float* a_s   = h   + (size_t)GAT_N * GAT_HC;          // N*8
    float* a_d   = a_s + (size_t)GAT_N * GAT_H;           // N*8
    float* score = a_d + (size_t)GAT_N * GAT_H;           // E*8
    float* acc   = score + (size_t)GAT_E * GAT_H;         // N*2048
    float* denom = acc + (size_t)GAT_N * GAT_HC;          // N*8
    unsigned* menc = (unsigned*)(denom + (size_t)GAT_N * GAT_H); // N*8 (u32)

    // zero acc + denom + menc in one capture-safe memset
    // (menc == 0 decodes to an order-minimal value, valid -inf substitute)
    size_t zero_bytes = ((size_t)GAT_N * GAT_HC + 2 * (size_t)GAT_N * GAT_H) * sizeof(float);
    hipMemsetAsync(acc, 0, zero_bytes, stream);

    // 1) GEMM: h = x @ W
    dim3 gemm_grid(GAT_HC / 64, (GAT_N + 127) / 128);
    gat_gemm_wmma<<<gemm_grid, 256, 0, stream>>>(x, W, h);

    // 2) per-node logits
    gat_logits<<<GAT_N, 256, 0, stream>>>(h, att_src, att_dst, a_s, a_d);

    // 3) edge scores + segment max
    gat_edge_score<<<(GAT_E + 255) / 256, 256, 0, stream>>>(ei, a_s, a_d, score, menc);

    // 4) exp + denom + weighted scatter
    gat_edge_scatter<<<GAT_E, 256, 0, stream>>>(ei, score, menc, h, acc, denom);

    // 5) normalize, mean heads, bias
    gat_finalize<<<GAT_N, 256, 0, stream>>>(acc, denom, bias, out);
}